// LDP_4964982194792
// MI455X (gfx1250) — compile-verified
//
#include <hip/hip_runtime.h>
#include <stdint.h>

// ---------------------------------------------------------------------------
// LDP alpha=0 census stencil, (B,512,512,3) f32 -> f32.
// Bandwidth-bound: ~201 MB min traffic -> ~8.6 us at 23.3 TB/s HBM.
// CDNA5 path: stage rows into LDS with GLOBAL_LOAD_ASYNC_TO_LDS_B128
// (ASYNCcnt) + s_wait_asynccnt, 12-tap stencil from LDS (conflict-free
// ds_load), non-temporal coalesced b32 stores (output is write-once; keep
// the 3x-reused input resident in the 192 MB L2 instead).
// ---------------------------------------------------------------------------

#ifndef __has_builtin
#define __has_builtin(x) 0
#endif

#if __has_builtin(__builtin_amdgcn_global_load_async_to_lds_b128)
#define LDP_HAVE_ASYNC_BUILTIN 1
#endif
#if __has_builtin(__builtin_amdgcn_s_wait_asynccnt)
#define LDP_HAVE_WAIT_BUILTIN 1
#endif

#define LDP_W     512
#define LDP_C     3
#define LDP_ROW   (LDP_W * LDP_C)                   // 1536 floats per image row
#define LDP_LPAD  8                                  // halo cols at [2..7]; [0..1] spare (16B align)
#define LDP_RPAD  4                                  // 3 halo floats + 1 spare
#define LDP_LROW  (LDP_LPAD + LDP_ROW + LDP_RPAD)   // 1548 floats = 6192 B (16B aligned)
#define LDP_BLOCK 256

// Builtin signature (probe-confirmed): (v4i32 addrspace(1)*, v4i32 addrspace(3)*, Imm, Imm)
typedef int ldp_v4i __attribute__((vector_size(4 * sizeof(int))));
typedef __attribute__((address_space(1))) ldp_v4i* ldp_gv4p;   // global v4i32*
typedef __attribute__((address_space(3))) ldp_v4i* ldp_lv4p;   // LDS v4i32*

__device__ __forceinline__ void ldp_async_b128(const float* gsrc, float* ldst) {
#ifdef LDP_HAVE_ASYNC_BUILTIN
  __builtin_amdgcn_global_load_async_to_lds_b128(
      (ldp_gv4p)(uintptr_t)gsrc,
      (ldp_lv4p)(unsigned)(uintptr_t)ldst,   // low 32 bits of generic = LDS offset
      /*imm offset*/ 0, /*cpol (RT: input is reused 3x)*/ 0);
#else
  unsigned           lds_off = (unsigned)(uintptr_t)ldst;
  unsigned long long ga      = (unsigned long long)(uintptr_t)gsrc;
  asm volatile("global_load_async_to_lds_b128 %0, %1, off"
               :: "v"(lds_off), "v"(ga)
               : "memory");
#endif
}

__device__ __forceinline__ void ldp_wait_async0() {
#ifdef LDP_HAVE_WAIT_BUILTIN
  __builtin_amdgcn_s_wait_asynccnt(0);
#else
  asm volatile("s_wait_asynccnt 0x0" ::: "memory");
#endif
}

__global__ __launch_bounds__(LDP_BLOCK) void ldp0_kernel(const float* __restrict__ x,
                                                         float* __restrict__ out) {
  __shared__ float smem[3 * LDP_LROW];  // 18576 B: rows h-1, h, h+1 with zero halos

  const int tid = threadIdx.x;
  const int bh  = blockIdx.x;            // b*512 + h
  const int h   = bh & (LDP_W - 1);
  const size_t rowBase = (size_t)bh * LDP_ROW;

  // Stage 3 input rows into LDS via async DMA-to-LDS; zero-fill OOB rows.
  // 1536 floats/row = 384 b128 transfers = 256 (all lanes) + 128 (tid<128):
  // straight-line issue, no divergent loop.
  for (int r = 0; r < 3; ++r) {
    const int row = h - 1 + r;
    float* ldsRow = smem + r * LDP_LROW + LDP_LPAD;
    if (row >= 0 && row < LDP_W) {
      const float* src = x + rowBase + (size_t)(row - h) * LDP_ROW;
      const int k = tid * 4;
      ldp_async_b128(src + k, ldsRow + k);                       // 16B/lane, coalesced
      if (tid < (LDP_ROW / 4 - LDP_BLOCK))                       // remaining 128 transfers
        ldp_async_b128(src + LDP_BLOCK * 4 + k, ldsRow + LDP_BLOCK * 4 + k);
    } else {
#pragma unroll
      for (int k = tid; k < LDP_ROW; k += LDP_BLOCK)
        ldsRow[k] = 0.0f;
    }
  }
  // Zero the left/right column halos (zero padding of the reference).
  if (tid < 3 * (LDP_LPAD + LDP_RPAD)) {
    const int r   = tid / (LDP_LPAD + LDP_RPAD);
    const int p   = tid % (LDP_LPAD + LDP_RPAD);
    const int pos = (p < LDP_LPAD) ? p : (LDP_LPAD + LDP_ROW + (p - LDP_LPAD));
    smem[r * LDP_LROW + pos] = 0.0f;
  }

  ldp_wait_async0();   // own-wave async copies landed in LDS
  __syncthreads();     // all waves' copies + zero fills visible

  const float* r0 = smem;                  // row h-1
  const float* r1 = smem + LDP_LROW;       // row h
  const float* r2 = smem + 2 * LDP_LROW;   // row h+1
  float* orow = out + rowBase;

#pragma unroll
  for (int i = 0; i < LDP_ROW / LDP_BLOCK; ++i) {     // 6 elements per thread
    const int t    = tid + i * LDP_BLOCK;
    const int base = LDP_LPAD + t;                     // (w*3 + c) at dw=0
    // taps at dw = -2,-1,0,+1  ->  float offsets -6,-3,0,+3 (channel-interleaved)
    const float a0 = r0[base - 6], b0 = r0[base - 3], c0 = r0[base], d0 = r0[base + 3];
    const float a1 = r1[base - 6], b1 = r1[base - 3], c1 = r1[base], d1 = r1[base + 3];
    const float a2 = r2[base - 6], b2 = r2[base - 3], c2 = r2[base], d2 = r2[base + 3];

    const float pc = b1 - c1;   // y21 - y22
    float z = 0.0f;
    z += ((a0 - b0) * pc >= 0.0f) ? 0.0f : 128.0f;   // (y10,y11)
    z += ((b0 - c0) * pc >= 0.0f) ? 0.0f : 64.0f;    // (y11,y12)
    z += ((c0 - d0) * pc >= 0.0f) ? 0.0f : 32.0f;    // (y12,y13)
    z += ((a1 - b1) * pc >= 0.0f) ? 0.0f : 1.0f;     // (y20,y21)
    z += ((c1 - d1) * pc >= 0.0f) ? 0.0f : 16.0f;    // (y22,y23)
    z += ((a2 - b2) * pc >= 0.0f) ? 0.0f : 2.0f;     // (y30,y31)
    z += ((b2 - c2) * pc >= 0.0f) ? 0.0f : 4.0f;     // (y31,y32)
    z += ((c2 - d2) * pc >= 0.0f) ? 0.0f : 8.0f;     // (y32,y33)

    // Write-once output: non-temporal store keeps reused input lines in L2.
    __builtin_nontemporal_store(z / 255.0f, orow + t);
  }
}

extern "C" void kernel_launch(void* const* d_in, const int* in_sizes, int n_in,
                              void* d_out, int out_size, void* d_ws, size_t ws_size,
                              hipStream_t stream) {
  (void)n_in; (void)out_size; (void)d_ws; (void)ws_size;
  const float* x   = (const float*)d_in[0];
  float*       out = (float*)d_out;
  const int B = in_sizes[0] / (LDP_W * LDP_W * LDP_C);   // 32 for the reference shapes
  ldp0_kernel<<<dim3(B * LDP_W), dim3(LDP_BLOCK), 0, stream>>>(x, out);
}